// LIF_70763881169334
// MI455X (gfx1250) — compile-verified
//
#include <hip/hip_runtime.h>
#include <stdint.h>

// LIF constants from the reference
#define LIF_T 8
#define LIF_THRESH 1.0f
// TAU == 1.0, GAMA unused in forward

typedef float v4f __attribute__((ext_vector_type(4)));

// One thread owns 4 consecutive channels (one float4) across all T=8 timesteps.
// Data is streamed global -> LDS with CDNA5 async B128 loads (ASYNCcnt),
// consumed via ds_load_b128, and spikes written with non-temporal B128 stores.
__global__ __launch_bounds__(256) void lif_async_kernel(
    const float* __restrict__ x, float* __restrict__ out, int planeF4) {
  // 8 timestep slabs of 256 float4 each = 32 KB LDS per block
  __shared__ v4f tile[LIF_T][256];

  const int tid = blockIdx.x * 256 + threadIdx.x;           // [0, planeF4)
  const uint64_t xbase = (uint64_t)(uintptr_t)x;            // SGPR base (GVS mode)
  const unsigned planeBytes = (unsigned)planeF4 * 16u;      // 16 MB per t-plane
  const unsigned gbyte = (unsigned)tid * 16u;               // byte offset in plane

  // Issue 8 independent async global->LDS 128-bit loads (fire-and-forget).
#pragma unroll
  for (int t = 0; t < LIF_T; ++t) {
    unsigned voff  = gbyte + (unsigned)t * planeBytes;      // 32-bit VGPR offset
    unsigned laddr = (unsigned)(uintptr_t)&tile[t][threadIdx.x]; // LDS byte addr
    asm volatile("global_load_async_to_lds_b128 %0, %1, %2"
                 :: "v"(laddr), "v"(voff), "s"(xbase)
                 : "memory");
  }
  // Wait for all async transfers to land in LDS.
  asm volatile("s_wait_asynccnt 0" ::: "memory");

  // Each lane reads back exactly the bytes it requested -> no barrier needed.
  v4f v[LIF_T];
#pragma unroll
  for (int t = 0; t < LIF_T; ++t) v[t] = tile[t][threadIdx.x];

  // LIF scan in registers. TAU=1: mem accumulates; spike resets mem to 0.
  float mx = 0.f, my = 0.f, mz = 0.f, mw = 0.f;
  v4f* __restrict__ outv = (v4f*)out;
#pragma unroll
  for (int t = 0; t < LIF_T; ++t) {
    mx += v[t].x; my += v[t].y; mz += v[t].z; mw += v[t].w;
    v4f s;
    s.x = (mx >= LIF_THRESH) ? LIF_THRESH : 0.0f;
    s.y = (my >= LIF_THRESH) ? LIF_THRESH : 0.0f;
    s.z = (mz >= LIF_THRESH) ? LIF_THRESH : 0.0f;
    s.w = (mw >= LIF_THRESH) ? LIF_THRESH : 0.0f;
    mx = (mx >= LIF_THRESH) ? 0.0f : mx;
    my = (my >= LIF_THRESH) ? 0.0f : my;
    mz = (mz >= LIF_THRESH) ? 0.0f : mz;
    mw = (mw >= LIF_THRESH) ? 0.0f : mw;
    // Write-once output stream: non-temporal B128 store (th:NT), keep L2 clean.
    __builtin_nontemporal_store(s, outv + (size_t)t * (size_t)planeF4 + tid);
  }
}

extern "C" void kernel_launch(void* const* d_in, const int* in_sizes, int n_in,
                              void* d_out, int out_size, void* d_ws, size_t ws_size,
                              hipStream_t stream) {
  (void)n_in; (void)out_size; (void)d_ws; (void)ws_size;
  const float* x = (const float*)d_in[0];
  float* out = (float*)d_out;

  // in_sizes[0] = T*B*F = 33,554,432. One float4 per thread per plane.
  const int total   = in_sizes[0];
  const int planeF4 = total / (LIF_T * 4);   // 1,048,576 float4 per timestep
  const int blocks  = planeF4 / 256;         // 4096 blocks, exact cover

  lif_async_kernel<<<blocks, 256, 0, stream>>>(x, out, planeF4);
}